// CompactExpandModule_54846732370229
// MI455X (gfx1250) — compile-verified
//
#include <hip/hip_runtime.h>
#include <hip/hip_bf16.h>

// out[b,s,:] = x[b,s,:] if token_ids[b,s] in keep_token_ids else 0
// Memory-bound streaming kernel for MI455X (gfx1250, wave32).
// Strategy: bitmap membership + per-row skip (masked rows write zeros, no read),
// 4 rows per block for 4x memory-level parallelism, async global<->LDS
// staging (ASYNCcnt) with non-temporal cache policy end-to-end.

typedef float v4f __attribute__((ext_vector_type(4)));
typedef int   v4i __attribute__((vector_size(16)));   // matches builtin param type

#define BITMAP_BITS  (1 << 20)            // covers token values [0, 2^20)
#define BITMAP_WORDS (BITMAP_BITS / 32)   // 32768 words = 128 KB of workspace
#define RPB     4                          // rows per block
#define THREADS 256
#define CPOL_NT 1                          // gfx12+ CPol: TH[2:0]=1 (non-temporal)

// ---- gfx1250 async global<->LDS path (builtins confirmed present, round 1) ----
#if defined(__AMDGCN__) && \
    __has_builtin(__builtin_amdgcn_global_load_async_to_lds_b128) && \
    __has_builtin(__builtin_amdgcn_global_store_async_from_lds_b128) && \
    __has_builtin(__builtin_amdgcn_s_wait_asynccnt)
#define USE_ASYNC_LDS 1
#else
#define USE_ASYNC_LDS 0
#endif

#if USE_ASYNC_LDS
typedef __attribute__((address_space(1))) v4i* gptr_v4i;
typedef __attribute__((address_space(3))) v4i* lptr_v4i;
#endif

// -------------------------------- kernels ----------------------------------

__global__ void cem_bitmap_zero(unsigned int* __restrict__ bm, int words) {
    int i = blockIdx.x * blockDim.x + threadIdx.x;
    if (i < words) bm[i] = 0u;
}

__global__ void cem_bitmap_set(const int* __restrict__ keep, int n,
                               unsigned int* __restrict__ bm) {
    int i = blockIdx.x * blockDim.x + threadIdx.x;
    if (i < n) {
        // streaming prefetch ahead (gfx1250: global_prefetch_b8; OOB prefetch
        // is speculative and silently dropped)
        __builtin_prefetch(keep + i + 2048, 0, 0);
        unsigned int t = (unsigned int)keep[i];
        if (t < (unsigned int)BITMAP_BITS) {
            atomicOr(&bm[t >> 5], 1u << (t & 31u));
        }
    }
}

__device__ __forceinline__ bool cem_is_kept(int t,
                                            const unsigned int* __restrict__ bm,
                                            const int* __restrict__ keep,
                                            int n_keep) {
    unsigned int u = (unsigned int)t;
    if (bm != nullptr && u < (unsigned int)BITMAP_BITS) {
        return (bm[u >> 5] >> (u & 31u)) & 1u;
    }
    // Slow fallback: only for token values outside bitmap range, or if the
    // workspace was too small for the bitmap. Never taken with reference data.
    for (int i = 0; i < n_keep; ++i) {
        if (keep[i] == t) return true;
    }
    return false;
}

// RPB rows per block; 256 threads, each moves one 16B chunk per row per step.
// All RPB async loads are issued before one wait -> 4x in-flight bytes/wave.
__global__ void __launch_bounds__(THREADS)
cem_mask_rows(const float* __restrict__ x,
              const int* __restrict__ tok,
              const unsigned int* __restrict__ bm,
              const int* __restrict__ keep, int n_keep,
              float* __restrict__ out, int d4 /* = D/4 */, int n_tok) {
    const int tid = threadIdx.x;
    const long long row0 = (long long)blockIdx.x * RPB;

    bool valid[RPB], kept[RPB];
#pragma unroll
    for (int r = 0; r < RPB; ++r) {
        const long long row = row0 + r;
        valid[r] = row < (long long)n_tok;
        kept[r]  = valid[r] && cem_is_kept(tok[row], bm, keep, n_keep);
    }

#if USE_ASYNC_LDS
    __shared__ __align__(16) float tile[RPB][THREADS * 4];   // 16 KB
    for (int cb = 0; cb < d4; cb += THREADS) {
        const int  i  = cb + tid;
        const bool in = i < d4;
        // issue all independent async loads first (MLP), NT policy
#pragma unroll
        for (int r = 0; r < RPB; ++r) {
            if (kept[r] && in) {
                const v4f* src = (const v4f*)(x + (row0 + r) * (long long)d4 * 4);
                __builtin_amdgcn_global_load_async_to_lds_b128(
                    (gptr_v4i)(v4i*)(src + i),
                    (lptr_v4i)(v4i*)&tile[r][tid * 4], 0, CPOL_NT);
            }
        }
        __builtin_amdgcn_s_wait_asynccnt(0);
        // drain: async NT stores for kept rows, NT zero stores for masked rows
#pragma unroll
        for (int r = 0; r < RPB; ++r) {
            if (!valid[r] || !in) continue;
            v4f* dst = (v4f*)(out + (row0 + r) * (long long)d4 * 4);
            if (kept[r]) {
                __builtin_amdgcn_global_store_async_from_lds_b128(
                    (gptr_v4i)(v4i*)(dst + i),
                    (lptr_v4i)(v4i*)&tile[r][tid * 4], 0, CPOL_NT);
            } else {
                v4f z = {0.0f, 0.0f, 0.0f, 0.0f};
                __builtin_nontemporal_store(z, dst + i);
            }
        }
        __builtin_amdgcn_s_wait_asynccnt(0);   // LDS reusable next chunk
    }
#else
    // NT 128-bit register streaming with the same 4-row MLP structure
    for (int cb = 0; cb < d4; cb += THREADS) {
        const int  i  = cb + tid;
        const bool in = i < d4;
        v4f v[RPB];
#pragma unroll
        for (int r = 0; r < RPB; ++r) {
            v[r] = (v4f){0.0f, 0.0f, 0.0f, 0.0f};
            if (kept[r] && in) {
                const v4f* src = (const v4f*)(x + (row0 + r) * (long long)d4 * 4);
                v[r] = __builtin_nontemporal_load(src + i);
            }
        }
#pragma unroll
        for (int r = 0; r < RPB; ++r) {
            if (valid[r] && in) {
                v4f* dst = (v4f*)(out + (row0 + r) * (long long)d4 * 4);
                __builtin_nontemporal_store(v[r], dst + i);
            }
        }
    }
#endif
}

// Scalar-tail variant in case D is not a multiple of 4 (not hit for D=1024).
__global__ void __launch_bounds__(THREADS)
cem_mask_rows_scalar(const float* __restrict__ x,
                     const int* __restrict__ tok,
                     const unsigned int* __restrict__ bm,
                     const int* __restrict__ keep, int n_keep,
                     float* __restrict__ out, int D) {
    const long long row = blockIdx.x;
    const int t = tok[row];
    const bool kept = cem_is_kept(t, bm, keep, n_keep);
    const float* src = x + row * (long long)D;
    float* dstp = out + row * (long long)D;
    for (int i = threadIdx.x; i < D; i += blockDim.x) {
        float v = kept ? __builtin_nontemporal_load(src + i) : 0.0f;
        __builtin_nontemporal_store(v, dstp + i);
    }
}

// ------------------------------ host launcher -------------------------------

extern "C" void kernel_launch(void* const* d_in, const int* in_sizes, int n_in,
                              void* d_out, int out_size, void* d_ws, size_t ws_size,
                              hipStream_t stream) {
    const float* x    = (const float*)d_in[0];   // [B,S,D] float32
    const int*   tok  = (const int*)d_in[1];     // [B,S]   int32
    const int*   keep = (const int*)d_in[2];     // [N_KEEP] int32

    const int n_embed = in_sizes[0];             // B*S*D
    const int n_tok   = in_sizes[1];             // B*S
    const int n_keep  = in_sizes[2];             // N_KEEP
    const int D       = n_embed / n_tok;         // 1024

    float* out = (float*)d_out;

    unsigned int* bm = nullptr;
    if (ws_size >= (size_t)BITMAP_WORDS * sizeof(unsigned int)) {
        bm = (unsigned int*)d_ws;
        cem_bitmap_zero<<<(BITMAP_WORDS + 255) / 256, 256, 0, stream>>>(bm, BITMAP_WORDS);
        cem_bitmap_set<<<(n_keep + 255) / 256, 256, 0, stream>>>(keep, n_keep, bm);
    }

    if ((D & 3) == 0) {
        const int nblk = (n_tok + RPB - 1) / RPB;
        cem_mask_rows<<<nblk, THREADS, 0, stream>>>(x, tok, bm, keep, n_keep,
                                                    out, D / 4, n_tok);
    } else {
        cem_mask_rows_scalar<<<n_tok, THREADS, 0, stream>>>(x, tok, bm, keep,
                                                            n_keep, out, D);
    }
}